// MyModel_80728205296076
// MI455X (gfx1250) — compile-verified
//
#include <hip/hip_runtime.h>
#include <math.h>

// ---------------------------------------------------------------------------
// Event-graph PointNet for MI455X (gfx1250, wave32).
// Static dims (match reference): N=262144 nodes, E=2097152 edges, B=32 graphs.
// Stage1 pool: grid=4 -> 32x32 cells -> n1=32768. Stage2: grid=8 -> 16x16 -> n2=8192.
// Final pool: 16 -> 8x8 cells x 64ch = 4096 per graph.
// ---------------------------------------------------------------------------

typedef __attribute__((ext_vector_type(2))) float v2f;
typedef __attribute__((ext_vector_type(8))) float v8f;

#define NNODES   262144
#define NEDGES   2097152
#define NGRAPH   32
#define N1       32768     // 32 * 32*32
#define N2       8192      // 32 * 16*16

// ----------------------------- utility -------------------------------------

__global__ void fill_zero_kernel(float* __restrict__ p, int n) {
    int i = blockIdx.x * blockDim.x + threadIdx.x;
    if (i < n) p[i] = 0.0f;
}

__device__ __forceinline__ void amax_f32_nonneg(float* addr, float v) {
    // values are >= 0 (post-ReLU / post-max of ReLU): int compare == float compare
    atomicMax((int*)addr, __float_as_int(v));
}

// ----------------------------- edge conv (WMMA) -----------------------------
// msg = relu([h[src], pos_xy[src]-pos_xy[dst]] @ W + b); h_out[dst] = max(msg)
// IN_H = input node channels, K = IN_H+2, K_PAD = K rounded up to mult of 4.
// 16-edge tile per wave; f32 WMMA 16x16x4 over K; OUT_CH/16 accumulators.

template<int IN_H, int K_PAD, int OUT_CH>
__global__ __launch_bounds__(256) void conv_wmma_kernel(
    const float* __restrict__ hin, const float* __restrict__ pos,
    const int*   __restrict__ src, const int*   __restrict__ dst,
    const float* __restrict__ W,   const float* __restrict__ bias,
    float* __restrict__ hout, int nNodes, int nEdges)
{
    constexpr int WAVES = 8;
    constexpr int NT    = OUT_CH / 16;

    __shared__ float Wl[K_PAD][OUT_CH];
    __shared__ float bl[OUT_CH];
    __shared__ float feat[WAVES][16][K_PAD];
    __shared__ int   dstl[WAVES][16];

    const int tid = threadIdx.x;

    // preload (zero-padded) weights + bias into LDS
    for (int idx = tid; idx < K_PAD * OUT_CH; idx += 256) {
        int k = idx / OUT_CH, n = idx % OUT_CH;
        Wl[k][n] = (k < IN_H + 2) ? W[k * OUT_CH + n] : 0.0f;
    }
    if (tid < OUT_CH) bl[tid] = bias[tid];
    __syncthreads();

    const int lane = tid & 31;
    const int w    = tid >> 5;
    const int gw   = blockIdx.x * WAVES + w;
    const int totalWaves = gridDim.x * WAVES;
    const int numTiles   = (nEdges + 15) >> 4;
    const int iters      = (numTiles + totalWaves - 1) / totalWaves;

    const int e    = lane & 15;   // edge-in-tile (also N index within tile)
    const int part = lane >> 4;   // half-wave selector
    const int kb   = part * 2;    // K sub-offset per A/B layout

    for (int t = 0; t < iters; ++t) {
        const int tile = gw + t * totalWaves;
        const bool tv = (tile < numTiles);   // wave-uniform

        if (tv) {
            const int eid = tile * 16 + e;
            int s = 0, d = nNodes;
            if (eid < nEdges) { s = src[eid]; d = dst[eid]; }
            // gather node features (both half-waves split the channels)
            for (int k = part; k < IN_H; k += 2)
                feat[w][e][k] = hin[s * IN_H + k];
            if (part == 0) {
                int dc = d; if (dc > nNodes - 1) dc = nNodes - 1; if (dc < 0) dc = 0;
                feat[w][e][IN_H]     = pos[s * 3 + 0] - pos[dc * 3 + 0];
                feat[w][e][IN_H + 1] = pos[s * 3 + 1] - pos[dc * 3 + 1];
                #pragma unroll
                for (int k = IN_H + 2; k < K_PAD; ++k) feat[w][e][k] = 0.0f;
                dstl[w][e] = (eid < nEdges && d >= 0 && d < nNodes) ? d : -1;
            }
        }
        __syncthreads();

        if (tv) {
            v8f acc[NT];
            #pragma unroll
            for (int nt = 0; nt < NT; ++nt)
                #pragma unroll
                for (int r = 0; r < 8; ++r) acc[nt][r] = 0.0f;

            for (int k0 = 0; k0 < K_PAD; k0 += 4) {
                v2f a;
                a.x = feat[w][e][k0 + kb];
                a.y = feat[w][e][k0 + kb + 1];
                #pragma unroll
                for (int nt = 0; nt < NT; ++nt) {
                    v2f b;
                    b.x = Wl[k0 + kb][nt * 16 + e];
                    b.y = Wl[k0 + kb + 1][nt * 16 + e];
                    acc[nt] = __builtin_amdgcn_wmma_f32_16x16x4_f32(
                        false, a, false, b, (short)0, acc[nt], false, false);
                }
            }
            // bias + relu + atomic segment-max scatter
            #pragma unroll
            for (int nt = 0; nt < NT; ++nt) {
                const int n = nt * 16 + e;
                const float bv = bl[n];
                #pragma unroll
                for (int r = 0; r < 8; ++r) {
                    const int m = r + part * 8;          // edge row in tile
                    const int d = dstl[w][m];
                    if (d >= 0) {
                        float v = acc[nt][r] + bv;
                        v = v > 0.0f ? v : 0.0f;
                        amax_f32_nonneg(&hout[d * OUT_CH + n], v);
                    }
                }
            }
        }
        __syncthreads();
    }
}

// ----------------------------- voxel pooling --------------------------------

template<int CH>
__global__ void pool_assign_kernel(
    const float* __restrict__ h, const float* __restrict__ pos,
    int* __restrict__ cluster, float* __restrict__ hp,
    float* __restrict__ possum, float* __restrict__ cnt,
    int nOld, int npg, int gridsz, int cells)
{
    int i = blockIdx.x * blockDim.x + threadIdx.x;
    if (i >= nOld) return;
    const int b = i / npg;
    const float px = pos[i * 3], py = pos[i * 3 + 1], pt = pos[i * 3 + 2];
    int ix = (int)floorf(px / (float)gridsz);
    int iy = (int)floorf(py / (float)gridsz);
    ix = ix < 0 ? 0 : (ix > cells - 1 ? cells - 1 : ix);
    iy = iy < 0 ? 0 : (iy > cells - 1 ? cells - 1 : iy);
    const int cl = b * cells * cells + iy * cells + ix;
    cluster[i] = cl;
    #pragma unroll
    for (int c = 0; c < CH; ++c)
        amax_f32_nonneg(&hp[cl * CH + c], h[i * CH + c]);
    atomicAdd(&possum[cl * 3 + 0], px);
    atomicAdd(&possum[cl * 3 + 1], py);
    atomicAdd(&possum[cl * 3 + 2], pt);
    atomicAdd(&cnt[cl], 1.0f);
}

__global__ void pos_div_kernel(const float* __restrict__ possum,
                               const float* __restrict__ cnt,
                               float* __restrict__ posnew, int nNew)
{
    int i = blockIdx.x * blockDim.x + threadIdx.x;
    if (i >= nNew) return;
    float c = cnt[i]; c = c > 1.0f ? c : 1.0f;
    posnew[i * 3 + 0] = possum[i * 3 + 0] / c;
    posnew[i * 3 + 1] = possum[i * 3 + 1] / c;
    posnew[i * 3 + 2] = possum[i * 3 + 2] / c;
}

__global__ void edge_remap_kernel(const int* __restrict__ src,
                                  const int* __restrict__ dst,
                                  const int* __restrict__ cluster,
                                  int* __restrict__ srcN, int* __restrict__ dstN,
                                  int nOld, int nNew, int nE)
{
    int e = blockIdx.x * blockDim.x + threadIdx.x;
    if (e >= nE) return;
    const int s = src[e], d = dst[e];
    const int sn = cluster[s];
    int dn;
    if (d < nOld) {
        int dc = d < 0 ? 0 : d;
        dn = cluster[dc];
    } else {
        dn = nNew;               // sentinel carried through
    }
    if (sn == dn) dn = nNew;     // drop self-loop
    srcN[e] = sn;
    dstN[e] = dn;
}

// ----------------------------- output grid pool ------------------------------
// nodes at stage 2: 256 per graph; pool to 8x8 cells x 64 ch -> g[B][4096]

__global__ void pool_out_kernel(const float* __restrict__ h,
                                const float* __restrict__ pos,
                                float* __restrict__ g, int nNodes)
{
    int i = blockIdx.x * blockDim.x + threadIdx.x;
    if (i >= nNodes) return;
    const int b = i >> 8;
    const float px = pos[i * 3], py = pos[i * 3 + 1];
    int ix = (int)floorf(px / 16.0f);
    int iy = (int)floorf(py / 16.0f);
    ix = ix < 0 ? 0 : (ix > 7 ? 7 : ix);
    iy = iy < 0 ? 0 : (iy > 7 ? 7 : iy);
    const int cl = iy * 8 + ix;
    const float* hr = h + (size_t)i * 64;
    float* gr = g + (size_t)b * 4096 + cl * 64;
    #pragma unroll
    for (int c = 0; c < 64; ++c)
        amax_f32_nonneg(&gr[c], hr[c]);
}

// ----------------------------- MLP head --------------------------------------
// fc1: [32,4096] @ [4096,1024] + b, relu  -> hidden  (f32 WMMA, 1 wave / 16x16 tile)

__global__ __launch_bounds__(128) void fc1_wmma_kernel(
    const float* __restrict__ g, const float* __restrict__ W,
    const float* __restrict__ bias, float* __restrict__ out)
{
    const int lane  = threadIdx.x & 31;
    const int w     = threadIdx.x >> 5;
    const int tile  = blockIdx.x * 4 + w;   // 32 blocks * 4 waves = 128 tiles
    const int mtile = tile & 1;             // 32 rows  -> 2 M tiles
    const int ntile = tile >> 1;            // 1024 cols -> 64 N tiles
    const int l     = lane & 15;
    const int part  = lane >> 4;
    const int kb    = part * 2;

    v8f acc;
    #pragma unroll
    for (int r = 0; r < 8; ++r) acc[r] = 0.0f;

    const float* grow = g + (size_t)(mtile * 16 + l) * 4096;
    const float* wcol = W + ntile * 16 + l;
    for (int k0 = 0; k0 < 4096; k0 += 4) {
        v2f a;
        a.x = grow[k0 + kb];
        a.y = grow[k0 + kb + 1];
        v2f b;
        b.x = wcol[(size_t)(k0 + kb) * 1024];
        b.y = wcol[(size_t)(k0 + kb + 1) * 1024];
        acc = __builtin_amdgcn_wmma_f32_16x16x4_f32(
            false, a, false, b, (short)0, acc, false, false);
    }
    const int n = ntile * 16 + l;
    const float bv = bias[n];
    #pragma unroll
    for (int r = 0; r < 8; ++r) {
        const int m = mtile * 16 + r + part * 8;
        float v = acc[r] + bv;
        v = v > 0.0f ? v : 0.0f;
        out[(size_t)m * 1024 + n] = v;
    }
}

// fc2: [32,1024] @ [1024,10] + b, then row-wise log_softmax -> out[32,10]

__global__ __launch_bounds__(320) void fc2_softmax_kernel(
    const float* __restrict__ hidden, const float* __restrict__ W,
    const float* __restrict__ bias, float* __restrict__ out)
{
    __shared__ float logits[320];
    const int t = threadIdx.x;          // 0..319
    const int b = t / 10, c = t % 10;
    float s = bias[c];
    const float* hr = hidden + (size_t)b * 1024;
    for (int k = 0; k < 1024; ++k) s += hr[k] * W[k * 10 + c];
    logits[t] = s;
    __syncthreads();
    float mx = -INFINITY;
    for (int j = 0; j < 10; ++j) mx = fmaxf(mx, logits[b * 10 + j]);
    float se = 0.0f;
    for (int j = 0; j < 10; ++j) se += expf(logits[b * 10 + j] - mx);
    out[b * 10 + c] = s - mx - logf(se);
}

// ----------------------------- host launch -----------------------------------

static inline void zero_f32(float* p, size_t n, hipStream_t s) {
    fill_zero_kernel<<<(unsigned)((n + 255) / 256), 256, 0, s>>>(p, (int)n);
}

extern "C" void kernel_launch(void* const* d_in, const int* in_sizes, int n_in,
                              void* d_out, int out_size, void* d_ws, size_t ws_size,
                              hipStream_t stream) {
    (void)in_sizes; (void)n_in; (void)out_size; (void)ws_size;

    const float* x    = (const float*)d_in[0];
    const float* pos0 = (const float*)d_in[1];
    const int*   ei   = (const int*)d_in[2];
    const int*   src0 = ei;
    const int*   dst0 = ei + NEDGES;
    const float* W1  = (const float*)d_in[4],  *b1  = (const float*)d_in[5];
    const float* W2  = (const float*)d_in[6],  *b2  = (const float*)d_in[7];
    const float* W3  = (const float*)d_in[8],  *b3  = (const float*)d_in[9];
    const float* W4  = (const float*)d_in[10], *b4  = (const float*)d_in[11];
    const float* W5  = (const float*)d_in[12], *b5  = (const float*)d_in[13];
    const float* Wl1 = (const float*)d_in[14], *bl1 = (const float*)d_in[15];
    const float* Wl2 = (const float*)d_in[16], *bl2 = (const float*)d_in[17];
    float* outp = (float*)d_out;

    // ---- workspace layout (bytes, 256B aligned) ----
    char* ws = (char*)d_ws;
    size_t o = 0;
    auto alloc = [&](size_t bytes) { char* p = ws + o; o += (bytes + 255) & ~size_t(255); return p; };
    float* h1      = (float*)alloc((size_t)NNODES * 16 * 4);
    int*   cl1     = (int*)  alloc((size_t)NNODES * 4);
    float* h1p     = (float*)alloc((size_t)N1 * 16 * 4);
    float* possum1 = (float*)alloc((size_t)N1 * 3 * 4);
    float* cnt1    = (float*)alloc((size_t)N1 * 4);
    float* pos1    = (float*)alloc((size_t)N1 * 3 * 4);
    int*   src1    = (int*)  alloc((size_t)NEDGES * 4);
    int*   dst1    = (int*)  alloc((size_t)NEDGES * 4);
    float* h2      = (float*)alloc((size_t)N1 * 32 * 4);
    float* h3      = (float*)alloc((size_t)N1 * 32 * 4);
    int*   cl2     = (int*)  alloc((size_t)N1 * 4);
    float* h3p     = (float*)alloc((size_t)N2 * 32 * 4);
    float* possum2 = (float*)alloc((size_t)N2 * 3 * 4);
    float* cnt2    = (float*)alloc((size_t)N2 * 4);
    float* pos2    = (float*)alloc((size_t)N2 * 3 * 4);
    int*   src2    = (int*)  alloc((size_t)NEDGES * 4);
    int*   dst2    = (int*)  alloc((size_t)NEDGES * 4);
    float* h4      = (float*)alloc((size_t)N2 * 64 * 4);
    float* h5      = (float*)alloc((size_t)N2 * 64 * 4);
    float* gbuf    = (float*)alloc((size_t)NGRAPH * 4096 * 4);
    float* hid     = (float*)alloc((size_t)NGRAPH * 1024 * 4);

    const int CONV_BLOCKS = 512;   // 512 blocks * 8 waves, grid-stride over 131072 tiles

    // conv1: [x(1), dpos(2)] -> 16ch  (K=3 -> pad 4)
    zero_f32(h1, (size_t)NNODES * 16, stream);
    conv_wmma_kernel<1, 4, 16><<<CONV_BLOCKS, 256, 0, stream>>>(
        x, pos0, src0, dst0, W1, b1, h1, NNODES, NEDGES);

    // pool1: grid 4 -> 32x32 cells
    zero_f32(h1p, (size_t)N1 * 16, stream);
    zero_f32(possum1, (size_t)N1 * 3, stream);
    zero_f32(cnt1, (size_t)N1, stream);
    pool_assign_kernel<16><<<(NNODES + 255) / 256, 256, 0, stream>>>(
        h1, pos0, cl1, h1p, possum1, cnt1, NNODES, 8192, 4, 32);
    pos_div_kernel<<<(N1 + 255) / 256, 256, 0, stream>>>(possum1, cnt1, pos1, N1);
    edge_remap_kernel<<<(NEDGES + 255) / 256, 256, 0, stream>>>(
        src0, dst0, cl1, src1, dst1, NNODES, N1, NEDGES);

    // conv2: [16, dpos] -> 32  (K=18 -> pad 20)
    zero_f32(h2, (size_t)N1 * 32, stream);
    conv_wmma_kernel<16, 20, 32><<<CONV_BLOCKS, 256, 0, stream>>>(
        h1p, pos1, src1, dst1, W2, b2, h2, N1, NEDGES);

    // conv3: [32, dpos] -> 32  (K=34 -> pad 36)
    zero_f32(h3, (size_t)N1 * 32, stream);
    conv_wmma_kernel<32, 36, 32><<<CONV_BLOCKS, 256, 0, stream>>>(
        h2, pos1, src1, dst1, W3, b3, h3, N1, NEDGES);

    // pool2: grid 8 -> 16x16 cells
    zero_f32(h3p, (size_t)N2 * 32, stream);
    zero_f32(possum2, (size_t)N2 * 3, stream);
    zero_f32(cnt2, (size_t)N2, stream);
    pool_assign_kernel<32><<<(N1 + 255) / 256, 256, 0, stream>>>(
        h3, pos1, cl2, h3p, possum2, cnt2, N1, 1024, 8, 16);
    pos_div_kernel<<<(N2 + 255) / 256, 256, 0, stream>>>(possum2, cnt2, pos2, N2);
    edge_remap_kernel<<<(NEDGES + 255) / 256, 256, 0, stream>>>(
        src1, dst1, cl2, src2, dst2, N1, N2, NEDGES);

    // conv4: [32, dpos] -> 64  (K=34 -> pad 36)
    zero_f32(h4, (size_t)N2 * 64, stream);
    conv_wmma_kernel<32, 36, 64><<<CONV_BLOCKS, 256, 0, stream>>>(
        h3p, pos2, src2, dst2, W4, b4, h4, N2, NEDGES);

    // conv5: [64, dpos] -> 64  (K=66 -> pad 68)
    zero_f32(h5, (size_t)N2 * 64, stream);
    conv_wmma_kernel<64, 68, 64><<<CONV_BLOCKS, 256, 0, stream>>>(
        h4, pos2, src2, dst2, W5, b5, h5, N2, NEDGES);

    // grid pool -> g[32, 4096]
    zero_f32(gbuf, (size_t)NGRAPH * 4096, stream);
    pool_out_kernel<<<(N2 + 255) / 256, 256, 0, stream>>>(h5, pos2, gbuf, N2);

    // MLP head
    fc1_wmma_kernel<<<32, 128, 0, stream>>>(gbuf, Wl1, bl1, hid);
    fc2_softmax_kernel<<<1, 320, 0, stream>>>(hid, Wl2, bl2, outp);
}